// CustomRNN_21706764714807
// MI455X (gfx1250) — compile-verified
//
#include <hip/hip_runtime.h>
#include <math.h>

// B=256, T=512, D=128, U=128, C=64
#define BB 256
#define TT 512
#define DD 128
#define UU 128
#define CC 64
#define MM (BB * TT)   // 131072 flattened (B*T) rows
#define S128 132       // LDS row stride (floats) for 128-wide tiles; 132%64=4, 132*4%16=0
#define S64  68        // LDS row stride (floats) for 64-wide tiles;  68%64=4,  68*4%16=0

typedef __attribute__((ext_vector_type(2))) float v2f;
typedef __attribute__((ext_vector_type(8))) float v8f;

// Exact-fp32 WMMA: D(16x16,f32) = A(16x4,f32) * B(4x16,f32) + C
__device__ __forceinline__ v8f wmma_f32_k4(v2f a, v2f b, v8f c) {
    return __builtin_amdgcn_wmma_f32_16x16x4_f32(false, a, false, b, (short)0, c,
                                                 false, false);
}

// --- gfx1250 async DMA helpers (ASYNCcnt-tracked), via inline asm ------------
__device__ __forceinline__ unsigned lds_off(const void* p) {
    // Generic VA of an LDS object: low 32 bits are the LDS byte address.
    return (unsigned)(uintptr_t)p;
}
__device__ __forceinline__ void async_g2l_b128(unsigned lds_addr, const void* gaddr) {
    asm volatile("global_load_async_to_lds_b128 %0, %1, off"
                 :: "v"(lds_addr), "v"((unsigned long long)(uintptr_t)gaddr)
                 : "memory");
}
__device__ __forceinline__ void async_l2g_b128(void* gaddr, unsigned lds_addr) {
    asm volatile("global_store_async_from_lds_b128 %0, %1, off"
                 :: "v"((unsigned long long)(uintptr_t)gaddr), "v"(lds_addr)
                 : "memory");
}
__device__ __forceinline__ void wait_async0() {
    asm volatile("s_wait_asynccnt 0x0" ::: "memory");
}

// ---------------------------------------------------------------------------
// Kernel 1: H = tanh(X @ W1 + b1)   X:(M,128), W1:(128,128), H:(M,128)
// 8 waves/block; block owns a 16-row strip. X tile staged by async DMA into
// LDS; result tile written to LDS and DMA'd out coalesced.
// ---------------------------------------------------------------------------
__global__ void __launch_bounds__(256)
k_in_proj(const float* __restrict__ X, const float* __restrict__ W1,
          const float* __restrict__ b1, float* __restrict__ H) {
    __shared__ float Xt[16 * S128];
    __shared__ float Ot[16 * S128];

    const int tid  = threadIdx.x;
    const int lane = tid & 31;
    const int wave = tid >> 5;           // 0..7 -> N tile
    const int r    = lane & 15;
    const int half = lane >> 4;
    const int m0   = blockIdx.x * 16;
    const int n0   = wave * 16;

    // Stage X tile (16x128 = 512 x 16B chunks) into LDS: 2 chunks per thread.
    {
        const unsigned xb = lds_off(&Xt[0]);
#pragma unroll
        for (int j = 0; j < 2; ++j) {
            const int q = tid + j * 256;           // chunk id
            const int row = q >> 5, c = q & 31;    // c: 16B chunk within row
            async_g2l_b128(xb + (unsigned)(row * S128 + c * 4) * 4u,
                           X + (size_t)(m0 + row) * DD + c * 4);
        }
        wait_async0();
    }
    __syncthreads();

    v8f a0 = {}, a1 = {}, a2 = {}, a3 = {};
    const float* w1p = W1 + n0 + r;
#pragma unroll
    for (int kk = 0; kk < 32; ++kk) {
        const int c = kk * 4 + 2 * half;
        v2f a, b;
        a.x = Xt[r * S128 + c];
        a.y = Xt[r * S128 + c + 1];
        b.x = w1p[(size_t)c * UU];
        b.y = w1p[(size_t)(c + 1) * UU];
        switch (kk & 3) {
            case 0: a0 = wmma_f32_k4(a, b, a0); break;
            case 1: a1 = wmma_f32_k4(a, b, a1); break;
            case 2: a2 = wmma_f32_k4(a, b, a2); break;
            default: a3 = wmma_f32_k4(a, b, a3); break;
        }
    }
    const v8f acc = (a0 + a1) + (a2 + a3);
    const float bias = b1[n0 + r];
#pragma unroll
    for (int i = 0; i < 8; ++i)
        Ot[(i + 8 * half) * S128 + n0 + r] = tanhf(acc[i] + bias);
    __syncthreads();

    // DMA the finished tile out, coalesced.
    {
        const unsigned ob = lds_off(&Ot[0]);
#pragma unroll
        for (int j = 0; j < 2; ++j) {
            const int q = tid + j * 256;
            const int row = q >> 5, c = q & 31;
            async_l2g_b128(H + (size_t)(m0 + row) * UU + c * 4,
                           ob + (unsigned)(row * S128 + c * 4) * 4u);
        }
        wait_async0();
    }
}

// ---------------------------------------------------------------------------
// Kernel 2: recurrence  state_t = h_t + tanh(state_{t-1} @ W2 + b2)
// 16 WGs x 16 batch rows. 8 waves; wave w owns columns [16w,16w+16).
// W2 fragments resident in VGPRs; state in LDS; h-tiles double-buffered in
// LDS with async DMA prefetch of step t+1 overlapping step t's WMMAs.
// Features overwrite H in place.
// ---------------------------------------------------------------------------
__global__ void __launch_bounds__(256)
k_recurrence(float* __restrict__ HF, const float* __restrict__ W2,
             const float* __restrict__ b2) {
    __shared__ float Sld[16 * S128];
    __shared__ float Ht[2][16 * S128];

    const int tid  = threadIdx.x;
    const int lane = tid & 31;
    const int wave = tid >> 5;
    const int r    = lane & 15;
    const int half = lane >> 4;
    const int n0   = wave * 16;
    const int b0   = blockIdx.x * 16;   // batch-row base

    // W2 column-tile as 32 B-fragments (64 VGPRs), resident for all 512 steps.
    v2f wf[32];
#pragma unroll
    for (int kk = 0; kk < 32; ++kk) {
        const int c = kk * 4 + 2 * half;
        wf[kk].x = W2[(size_t)c * UU + n0 + r];
        wf[kk].y = W2[(size_t)(c + 1) * UU + n0 + r];
    }
    const float bias = b2[n0 + r];

    // state_0 = 0
    for (int idx = tid; idx < 16 * S128; idx += 256) Sld[idx] = 0.0f;

    // Per-wave DMA of one h-tile: wave w covers rows {w, w+8} (chunks tid, tid+256).
    const int row0 = tid >> 5, c0 = tid & 31;    // == (wave, lane)
    const int row1 = row0 + 8;
    // Prologue: stage h-tile for t=0.
    {
        const unsigned hb = lds_off(&Ht[0][0]);
        async_g2l_b128(hb + (unsigned)(row0 * S128 + c0 * 4) * 4u,
                       HF + ((size_t)(b0 + row0) * TT + 0) * UU + c0 * 4);
        async_g2l_b128(hb + (unsigned)(row1 * S128 + c0 * 4) * 4u,
                       HF + ((size_t)(b0 + row1) * TT + 0) * UU + c0 * 4);
        wait_async0();
    }
    __syncthreads();

    for (int t = 0; t < TT; ++t) {
        // Prefetch next step's h-tile into the other buffer (overlaps WMMAs).
        if (t + 1 < TT) {
            const unsigned hb = lds_off(&Ht[(t + 1) & 1][0]);
            async_g2l_b128(hb + (unsigned)(row0 * S128 + c0 * 4) * 4u,
                           HF + ((size_t)(b0 + row0) * TT + (t + 1)) * UU + c0 * 4);
            async_g2l_b128(hb + (unsigned)(row1 * S128 + c0 * 4) * 4u,
                           HF + ((size_t)(b0 + row1) * TT + (t + 1)) * UU + c0 * 4);
        }

        // acc = state @ W2 (this wave's 16 columns), K=128, 4-way accumulator split.
        v8f a0 = {}, a1 = {}, a2 = {}, a3 = {};
#pragma unroll
        for (int kk = 0; kk < 32; ++kk) {
            const int c = kk * 4 + 2 * half;
            v2f a;
            a.x = Sld[r * S128 + c];
            a.y = Sld[r * S128 + c + 1];
            switch (kk & 3) {
                case 0: a0 = wmma_f32_k4(a, wf[kk], a0); break;
                case 1: a1 = wmma_f32_k4(a, wf[kk], a1); break;
                case 2: a2 = wmma_f32_k4(a, wf[kk], a2); break;
                default: a3 = wmma_f32_k4(a, wf[kk], a3); break;
            }
        }
        const v8f acc = (a0 + a1) + (a2 + a3);
        __syncthreads();   // all waves done reading old state

        const float* hbuf = &Ht[t & 1][0];
#pragma unroll
        for (int i = 0; i < 8; ++i) {
            const int mrow = i + 8 * half;
            const float h = hbuf[mrow * S128 + n0 + r];
            const float s = h + tanhf(acc[i] + bias);
            HF[((size_t)(b0 + mrow) * TT + t) * UU + n0 + r] = s;  // feature (in place)
            Sld[mrow * S128 + n0 + r] = s;                         // new state
        }
        wait_async0();     // my t+1 DMA chunks have landed
        __syncthreads();   // everyone's DMA + state writes visible
    }
}

// ---------------------------------------------------------------------------
// Kernel 3: OUT = F @ Wc + bc   F:(M,128), Wc:(128,64), OUT:(M,64)
// 4 waves/block; F tile DMA-staged in, OUT tile DMA-staged out.
// ---------------------------------------------------------------------------
__global__ void __launch_bounds__(128)
k_head(const float* __restrict__ F, const float* __restrict__ Wc,
       const float* __restrict__ bc, float* __restrict__ OUT) {
    __shared__ float Ft[16 * S128];
    __shared__ float Ot[16 * S64];

    const int tid  = threadIdx.x;
    const int lane = tid & 31;
    const int wave = tid >> 5;           // 0..3
    const int r    = lane & 15;
    const int half = lane >> 4;
    const int m0   = blockIdx.x * 16;
    const int n0   = wave * 16;

    {
        const unsigned fb = lds_off(&Ft[0]);
#pragma unroll
        for (int j = 0; j < 4; ++j) {
            const int q = tid + j * 128;          // 512 chunks, 4 per thread
            const int row = q >> 5, c = q & 31;
            async_g2l_b128(fb + (unsigned)(row * S128 + c * 4) * 4u,
                           F + (size_t)(m0 + row) * UU + c * 4);
        }
        wait_async0();
    }
    __syncthreads();

    v8f a0 = {}, a1 = {}, a2 = {}, a3 = {};
    const float* wcp = Wc + n0 + r;
#pragma unroll
    for (int kk = 0; kk < 32; ++kk) {
        const int c = kk * 4 + 2 * half;
        v2f a, b;
        a.x = Ft[r * S128 + c];
        a.y = Ft[r * S128 + c + 1];
        b.x = wcp[(size_t)c * CC];
        b.y = wcp[(size_t)(c + 1) * CC];
        switch (kk & 3) {
            case 0: a0 = wmma_f32_k4(a, b, a0); break;
            case 1: a1 = wmma_f32_k4(a, b, a1); break;
            case 2: a2 = wmma_f32_k4(a, b, a2); break;
            default: a3 = wmma_f32_k4(a, b, a3); break;
        }
    }
    const v8f acc = (a0 + a1) + (a2 + a3);
    const float bias = bc[n0 + r];
#pragma unroll
    for (int i = 0; i < 8; ++i)
        Ot[(i + 8 * half) * S64 + n0 + r] = acc[i] + bias;
    __syncthreads();

    {
        const unsigned ob = lds_off(&Ot[0]);
#pragma unroll
        for (int j = 0; j < 2; ++j) {
            const int q = tid + j * 128;          // 256 x 16B chunks
            const int row = q >> 4, c = q & 15;
            async_l2g_b128(OUT + (size_t)(m0 + row) * CC + c * 4,
                           ob + (unsigned)(row * S64 + c * 4) * 4u);
        }
        wait_async0();
    }
}

// ---------------------------------------------------------------------------
extern "C" void kernel_launch(void* const* d_in, const int* in_sizes, int n_in,
                              void* d_out, int out_size, void* d_ws, size_t ws_size,
                              hipStream_t stream) {
    const float* X  = (const float*)d_in[0];  // (B,T,D)
    const float* W1 = (const float*)d_in[1];  // (D,U)
    const float* b1 = (const float*)d_in[2];  // (U,)
    const float* W2 = (const float*)d_in[3];  // (U,U)
    const float* b2 = (const float*)d_in[4];  // (U,)
    const float* Wc = (const float*)d_in[5];  // (U,C)
    const float* bc = (const float*)d_in[6];  // (C,)
    float* OUT = (float*)d_out;               // (B,T,C) f32
    float* H   = (float*)d_ws;                // (B,T,U) f32: H then features

    (void)in_sizes; (void)n_in; (void)out_size; (void)ws_size;

    k_in_proj<<<MM / 16, 256, 0, stream>>>(X, W1, b1, H);
    k_recurrence<<<BB / 16, 256, 0, stream>>>(H, W2, b2);
    k_head<<<MM / 16, 128, 0, stream>>>(H, Wc, bc, OUT);
}